// GraphTransformer_13159779794966
// MI455X (gfx1250) — compile-verified
//
#include <hip/hip_runtime.h>
#include <math.h>

#define DIMC 128
#define HEADS 8
#define DHEAD 64
#define INNER 512
#define FFD 512
#define NLAYERS 4
#define NBEAD 256
#define INNODE 260
#define BATCH 2
#define NNODE 256
#define ATT_SCALE 0.125f  /* 64^-0.5 */

typedef __attribute__((ext_vector_type(2))) float v2f;
typedef __attribute__((ext_vector_type(8))) float v8f;

static __device__ __forceinline__ v8f wmma4(v2f a, v2f b, v8f c) {
  return __builtin_amdgcn_wmma_f32_16x16x4_f32(false, a, false, b, (short)0, c,
                                               false, false);
}

// ---------------- preprocessing ----------------

__global__ void center_kernel(const float* __restrict__ x, float* __restrict__ xc) {
  int b = blockIdx.x;
  int n = threadIdx.x;  // 256
  __shared__ float sm[3][256];
  float v0 = x[(b * NNODE + n) * 3 + 0];
  float v1 = x[(b * NNODE + n) * 3 + 1];
  float v2 = x[(b * NNODE + n) * 3 + 2];
  sm[0][n] = v0; sm[1][n] = v1; sm[2][n] = v2;
  __syncthreads();
  for (int s = 128; s > 0; s >>= 1) {
    if (n < s) { sm[0][n] += sm[0][n + s]; sm[1][n] += sm[1][n + s]; sm[2][n] += sm[2][n + s]; }
    __syncthreads();
  }
  float m0 = sm[0][0] * (1.0f / NNODE);
  float m1 = sm[1][0] * (1.0f / NNODE);
  float m2 = sm[2][0] * (1.0f / NNODE);
  xc[(b * NNODE + n) * 3 + 0] = v0 - m0;
  xc[(b * NNODE + n) * 3 + 1] = v1 - m1;
  xc[(b * NNODE + n) * 3 + 2] = v2 - m2;
}

__global__ void node_embed_kernel(const float* __restrict__ h, const float* __restrict__ xc,
                                  const float* __restrict__ t, const float* __restrict__ Wn,
                                  const float* __restrict__ bn, float* __restrict__ nodes) {
  int row = blockIdx.x;               // b*256 + n
  int b = row / NNODE, n = row % NNODE;
  int c = threadIdx.x;                // 128
  float acc = bn[c];
  for (int k = 0; k < NBEAD; k++) acc += h[n * NBEAD + k] * Wn[k * DIMC + c];
  for (int k = 0; k < 3; k++)     acc += xc[row * 3 + k] * Wn[(NBEAD + k) * DIMC + c];
  acc += t[b] * Wn[(NBEAD + 3) * DIMC + c];
  nodes[row * DIMC + c] = acc;
}

__global__ void dist_kernel(const float* __restrict__ xc, float* __restrict__ dist) {
  int bi = blockIdx.x;                // b*256 + i
  int j = threadIdx.x;                // 256
  int b = bi / NNODE;
  float d = 0.f;
  for (int k = 0; k < 3; k++) {
    float df = xc[bi * 3 + k] - xc[(b * NNODE + j) * 3 + k];
    d += df * df;
  }
  dist[(size_t)bi * NNODE + j] = d;
}

// u[l,hd] = W_edge . We[l,:,hd] ; v0[l,hd] = b_edge . We[l,:,hd] + be[l,hd]
__global__ void uv_kernel(const float* __restrict__ We, const float* __restrict__ be,
                          const float* __restrict__ Wedge, const float* __restrict__ bedge,
                          float* __restrict__ u, float* __restrict__ v0) {
  int l = blockIdx.x;
  int j = threadIdx.x;  // 512
  const float* W = We + (size_t)l * DIMC * INNER;
  float su = 0.f, sv = 0.f;
  for (int c = 0; c < DIMC; c++) {
    float w = W[c * INNER + j];
    su += Wedge[c] * w;
    sv += bedge[c] * w;
  }
  u[l * INNER + j] = su;
  v0[l * INNER + j] = sv + be[l * INNER + j];
}

// ---------------- layernorm / gate / decode ----------------

__global__ void ln_kernel(const float* __restrict__ xin, const float* __restrict__ g,
                          const float* __restrict__ b_, float* __restrict__ y) {
  int row = blockIdx.x;
  int c = threadIdx.x;  // 128
  __shared__ float sm[128];
  float v = xin[row * DIMC + c];
  sm[c] = v;
  __syncthreads();
  for (int s = 64; s > 0; s >>= 1) { if (c < s) sm[c] += sm[c + s]; __syncthreads(); }
  float mean = sm[0] * (1.0f / DIMC);
  __syncthreads();
  float d = v - mean;
  sm[c] = d * d;
  __syncthreads();
  for (int s = 64; s > 0; s >>= 1) { if (c < s) sm[c] += sm[c + s]; __syncthreads(); }
  float var = sm[0] * (1.0f / DIMC);
  y[row * DIMC + c] = d * rsqrtf(var + 1e-5f) * g[c] + b_[c];
}

__global__ void gate_kernel(const float* __restrict__ xnew, const float* __restrict__ Wg,
                            float* __restrict__ nodes) {
  int row = blockIdx.x;
  int c = threadIdx.x;  // 128
  __shared__ float sm[128];
  float xv = xnew[row * DIMC + c];
  float rv = nodes[row * DIMC + c];
  sm[c] = xv * Wg[c] + rv * Wg[DIMC + c] + (xv - rv) * Wg[2 * DIMC + c];
  __syncthreads();
  for (int s = 64; s > 0; s >>= 1) { if (c < s) sm[c] += sm[c + s]; __syncthreads(); }
  float g = 1.0f / (1.0f + expf(-sm[0]));
  nodes[row * DIMC + c] = xv * g + rv * (1.0f - g);
}

__global__ void dec_kernel(const float* __restrict__ nodes, const float* __restrict__ Wd,
                           const float* __restrict__ bd, float* __restrict__ out) {
  int idx = blockIdx.x * blockDim.x + threadIdx.x;
  if (idx >= BATCH * NNODE * 3) return;
  int k = idx % 3;
  int row = idx / 3;
  float acc = bd[k];
  for (int c = 0; c < DIMC; c++) acc += nodes[row * DIMC + c] * Wd[c * 3 + k];
  out[idx] = acc;
}

// ---------------- WMMA GEMM: C[M,N] = act(A[M,K] @ W[K,N] + bias) ----------------
// Compile-time N/K; one wave computes a 16x64 strip (4 n-tiles), so each
// A-fragment b64 load feeds 4 WMMAs and all addressing folds to immediates.

template <int N, int K, int ACT>
__global__ void __launch_bounds__(32)
gemm_kernel(const float* __restrict__ A, const float* __restrict__ W,
            const float* __restrict__ bias, float* __restrict__ C) {
  int lane = threadIdx.x;
  int l16 = lane & 15, half = lane >> 4;
  int n0 = blockIdx.x * 64;          // strip of 4 n-tiles
  int m0 = blockIdx.y * 16;
  v8f acc[4];
  v8f zero = {};
#pragma unroll
  for (int t = 0; t < 4; t++) acc[t] = zero;

  const float* arow = A + (size_t)(m0 + l16) * K + 2 * half;
  const float* wrow = W + (size_t)(2 * half) * N + n0 + l16;

#pragma unroll
  for (int k0 = 0; k0 < K; k0 += 4) {
    v2f a = *(const v2f*)(arow + k0);               // A[m][k0+2h .. +1]
    const float* w0 = wrow + (size_t)k0 * N;        // row k0+2h
#pragma unroll
    for (int t = 0; t < 4; t++) {
      v2f b;
      b.x = w0[16 * t];
      b.y = w0[N + 16 * t];                         // row k0+2h+1
      acc[t] = wmma4(a, b, acc[t]);
    }
  }
#pragma unroll
  for (int t = 0; t < 4; t++) {
    float bv = bias[n0 + 16 * t + l16];
#pragma unroll
    for (int r = 0; r < 8; r++) {
      int m = m0 + r + 8 * half;
      float v = acc[t][r] + bv;
      if (ACT == 1) v = 0.5f * v * (1.0f + erff(v * 0.70710678118654752f));
      C[(size_t)m * N + n0 + 16 * t + l16] = v;
    }
  }
}

// ---------------- alpha/beta: per (b,i,h) edge-bias scalars ----------------

__global__ void alphabeta_kernel(const float* __restrict__ q, const float* __restrict__ u,
                                 const float* __restrict__ v0, float* __restrict__ alpha,
                                 float* __restrict__ beta) {
  int idx = blockIdx.x * blockDim.x + threadIdx.x;  // b*256*8
  if (idx >= BATCH * NNODE * HEADS) return;
  int hh = idx % HEADS;
  int row = idx / HEADS;
  const float* qp = q + (size_t)row * INNER + hh * DHEAD;
  const float* up = u + hh * DHEAD;
  const float* vp = v0 + hh * DHEAD;
  float sa = 0.f, sb = 0.f;
  for (int d = 0; d < DHEAD; d++) { float qq = qp[d]; sa += qq * up[d]; sb += qq * vp[d]; }
  alpha[idx] = sa;
  beta[idx] = sb;
}

// ---------------- fused flash attention with rank-1 edge term ----------------
// one wave per (b, h, 16-query tile); logits = (q.k + dist*alpha + beta)*SCALE
// out = (P@V)/l + (s/l)*u + v0, s = sum_j P*dist

__global__ void __launch_bounds__(32)
attn_kernel(const float* __restrict__ q, const float* __restrict__ kv,
            const float* __restrict__ dist, const float* __restrict__ alpha,
            const float* __restrict__ beta, const float* __restrict__ u,
            const float* __restrict__ v0, float* __restrict__ out) {
  int blk = blockIdx.x;
  int it = blk & 15;
  int h = (blk >> 4) & 7;
  int b = blk >> 7;
  int i0 = it * 16;
  int lane = threadIdx.x, l16 = lane & 15, half = lane >> 4;

  __shared__ float p_lds[16 * 16];

  // preload Q A-fragments (16 k-steps over d=64) as b64 loads
  v2f qf[16];
  const float* qbase = q + (size_t)(b * NNODE + i0 + l16) * INNER + h * DHEAD + 2 * half;
#pragma unroll
  for (int kk = 0; kk < 16; kk++) qf[kk] = *(const v2f*)(qbase + 4 * kk);

  float alpha_r[8], beta_r[8];
#pragma unroll
  for (int r = 0; r < 8; r++) {
    int row = i0 + r + 8 * half;
    alpha_r[r] = alpha[(b * NNODE + row) * HEADS + h];
    beta_r[r] = beta[(b * NNODE + row) * HEADS + h];
  }
  float m_run[8], l_run[8], s_run[8];
#pragma unroll
  for (int r = 0; r < 8; r++) { m_run[r] = -INFINITY; l_run[r] = 0.f; s_run[r] = 0.f; }
  v8f zero = {};
  v8f o[4];
#pragma unroll
  for (int t = 0; t < 4; t++) o[t] = zero;

  const float* kbase = kv + (size_t)(b * NNODE) * (2 * INNER) + h * DHEAD;
  const float* vbase = kbase + INNER;
  const float* drow = dist + (size_t)(b * NNODE) * NNODE;

  for (int j0 = 0; j0 < NNODE; j0 += 16) {
    // S tile = Q @ K^T
    v8f s = zero;
    const float* kcol = kbase + (size_t)(j0 + l16) * (2 * INNER) + 2 * half;
#pragma unroll
    for (int kk = 0; kk < 16; kk++) {
      v2f bf = *(const v2f*)(kcol + 4 * kk);
      s = wmma4(qf[kk], bf, s);
    }
    // online softmax with edge bias
    float pcol[8];
#pragma unroll
    for (int r = 0; r < 8; r++) {
      int row = i0 + r + 8 * half;
      float dd = drow[(size_t)row * NNODE + j0 + l16];
      float logit = (s[r] + dd * alpha_r[r] + beta_r[r]) * ATT_SCALE;
      float tm = logit;
#pragma unroll
      for (int m = 8; m >= 1; m >>= 1) tm = fmaxf(tm, __shfl_xor(tm, m, 16));
      float m_new = fmaxf(m_run[r], tm);
      float sc = expf(m_run[r] - m_new);
      float p = expf(logit - m_new);
      float rs = p;
#pragma unroll
      for (int m = 8; m >= 1; m >>= 1) rs += __shfl_xor(rs, m, 16);
      float dsum = p * dd;
#pragma unroll
      for (int m = 8; m >= 1; m >>= 1) dsum += __shfl_xor(dsum, m, 16);
      l_run[r] = l_run[r] * sc + rs;
      s_run[r] = s_run[r] * sc + dsum;
      m_run[r] = m_new;
#pragma unroll
      for (int t = 0; t < 4; t++) o[t][r] *= sc;
      pcol[r] = p;
    }
    __syncthreads();
#pragma unroll
    for (int r = 0; r < 8; r++) p_lds[(r + 8 * half) * 16 + l16] = pcol[r];
    __syncthreads();
    // O += P @ V  (P via LDS layout swap: C-layout -> A-layout)
#pragma unroll
    for (int kk = 0; kk < 4; kk++) {
      v2f a = *(const v2f*)(&p_lds[l16 * 16 + 4 * kk + 2 * half]);
      int jr0 = j0 + 4 * kk + 2 * half;
      const float* vr0 = vbase + (size_t)jr0 * (2 * INNER) + l16;
      const float* vr1 = vr0 + 2 * INNER;
#pragma unroll
      for (int t = 0; t < 4; t++) {
        v2f bf;
        bf.x = vr0[16 * t];
        bf.y = vr1[16 * t];
        o[t] = wmma4(a, bf, o[t]);
      }
    }
    __syncthreads();
  }
  // epilogue: normalize + rank-1 edge value correction
#pragma unroll
  for (int t = 0; t < 4; t++) {
#pragma unroll
    for (int r = 0; r < 8; r++) {
      int row = i0 + r + 8 * half;
      float inv = 1.0f / l_run[r];
      int col = h * DHEAD + 16 * t + l16;
      float val = o[t][r] * inv + (s_run[r] * inv) * u[col] + v0[col];
      out[(size_t)(b * NNODE + row) * INNER + col] = val;
    }
  }
}

// ---------------- host orchestration ----------------

extern "C" void kernel_launch(void* const* d_in, const int* in_sizes, int n_in,
                              void* d_out, int out_size, void* d_ws, size_t ws_size,
                              hipStream_t stream) {
  (void)in_sizes; (void)n_in; (void)out_size; (void)ws_size;
  const float* x      = (const float*)d_in[0];
  const float* h      = (const float*)d_in[1];
  const float* t      = (const float*)d_in[2];
  const float* W_node = (const float*)d_in[3];
  const float* b_node = (const float*)d_in[4];
  const float* W_edge = (const float*)d_in[5];
  const float* b_edge = (const float*)d_in[6];
  const float* ln_a_g = (const float*)d_in[7];
  const float* ln_a_b = (const float*)d_in[8];
  const float* Wq     = (const float*)d_in[9];
  const float* bq     = (const float*)d_in[10];
  const float* Wkv    = (const float*)d_in[11];
  const float* bkv    = (const float*)d_in[12];
  const float* We     = (const float*)d_in[13];
  const float* be     = (const float*)d_in[14];
  const float* Wo     = (const float*)d_in[15];
  const float* bo     = (const float*)d_in[16];
  const float* Wg1    = (const float*)d_in[17];
  const float* ln_f_g = (const float*)d_in[18];
  const float* ln_f_b = (const float*)d_in[19];
  const float* W1     = (const float*)d_in[20];
  const float* b1     = (const float*)d_in[21];
  const float* W2     = (const float*)d_in[22];
  const float* b2     = (const float*)d_in[23];
  const float* Wg2    = (const float*)d_in[24];
  const float* W_dec  = (const float*)d_in[25];
  const float* b_dec  = (const float*)d_in[26];

  float* ws = (float*)d_ws;
  float* xc     = ws; ws += 1536;
  float* nodes  = ws; ws += 65536;
  float* xln    = ws; ws += 65536;
  float* qb     = ws; ws += 262144;
  float* kvb    = ws; ws += 524288;
  float* distb  = ws; ws += 131072;
  float* ub     = ws; ws += 2048;
  float* v0b    = ws; ws += 2048;
  float* alphab = ws; ws += 4096;
  float* betab  = ws; ws += 4096;
  float* attnb  = ws; ws += 262144;
  float* projb  = ws; ws += 65536;
  float* ffhb   = ws; ws += 262144;

  center_kernel<<<BATCH, NNODE, 0, stream>>>(x, xc);
  node_embed_kernel<<<BATCH * NNODE, DIMC, 0, stream>>>(h, xc, t, W_node, b_node, nodes);
  dist_kernel<<<BATCH * NNODE, NNODE, 0, stream>>>(xc, distb);
  uv_kernel<<<NLAYERS, INNER, 0, stream>>>(We, be, W_edge, b_edge, ub, v0b);

  const int M = BATCH * NNODE;  // 512
  for (int l = 0; l < NLAYERS; l++) {
    ln_kernel<<<M, DIMC, 0, stream>>>(nodes, ln_a_g + l * DIMC, ln_a_b + l * DIMC, xln);
    gemm_kernel<INNER, DIMC, 0><<<dim3(INNER / 64, M / 16), 32, 0, stream>>>(
        xln, Wq + (size_t)l * DIMC * INNER, bq + l * INNER, qb);
    gemm_kernel<2 * INNER, DIMC, 0><<<dim3(2 * INNER / 64, M / 16), 32, 0, stream>>>(
        xln, Wkv + (size_t)l * DIMC * 2 * INNER, bkv + l * 2 * INNER, kvb);
    alphabeta_kernel<<<16, 256, 0, stream>>>(qb, ub + l * INNER, v0b + l * INNER, alphab, betab);
    attn_kernel<<<BATCH * HEADS * (NNODE / 16), 32, 0, stream>>>(
        qb, kvb, distb, alphab, betab, ub + l * INNER, v0b + l * INNER, attnb);
    gemm_kernel<DIMC, INNER, 0><<<dim3(DIMC / 64, M / 16), 32, 0, stream>>>(
        attnb, Wo + (size_t)l * INNER * DIMC, bo + l * DIMC, projb);
    gate_kernel<<<M, DIMC, 0, stream>>>(projb, Wg1 + l * 3 * DIMC, nodes);
    ln_kernel<<<M, DIMC, 0, stream>>>(nodes, ln_f_g + l * DIMC, ln_f_b + l * DIMC, xln);
    gemm_kernel<FFD, DIMC, 1><<<dim3(FFD / 64, M / 16), 32, 0, stream>>>(
        xln, W1 + (size_t)l * DIMC * FFD, b1 + l * FFD, ffhb);
    gemm_kernel<DIMC, FFD, 0><<<dim3(DIMC / 64, M / 16), 32, 0, stream>>>(
        ffhb, W2 + (size_t)l * FFD * DIMC, b2 + l * DIMC, projb);
    gate_kernel<<<M, DIMC, 0, stream>>>(projb, Wg2 + l * 3 * DIMC, nodes);
  }
  dec_kernel<<<6, 256, 0, stream>>>(nodes, W_dec, b_dec, (float*)d_out);
}